// GATWrapper_12429635355066
// MI455X (gfx1250) — compile-verified
//
#include <hip/hip_runtime.h>
#include <hip/hip_bf16.h>
#include <math.h>

// ---------------------------------------------------------------------------
// GAT layer for MI455X (gfx1250, wave32).
//   h   = x @ emb_w + emb_b                     [N,128]
//   xh  = h @ gat_w                             [N,3,128]   <-- f32 WMMA GEMM
//   e   = leakyrelu(a_src[src]+a_dst[dst]); segment softmax over dst
//   agg = segment_sum(xh[src]*alpha)            [N,3,128]   <-- atomic scatter
//   out = (agg.mean(heads)+bias+h) @ out_w + out_b          [N,4]
// ---------------------------------------------------------------------------

#define HEADS 3
#define CH    128
#define HC    (HEADS * CH)   // 384
#define KDIM  128
#define NEG_SLOPE 0.2f
#define EPS 1e-16f

typedef float v2f __attribute__((ext_vector_type(2)));
typedef float v8f __attribute__((ext_vector_type(8)));

// Order-preserving float <-> uint mapping for atomic max on floats.
__device__ __forceinline__ unsigned f2ord(float f) {
    unsigned u = __float_as_uint(f);
    return (u & 0x80000000u) ? ~u : (u | 0x80000000u);
}
__device__ __forceinline__ float ord2f(unsigned u) {
    return __uint_as_float((u & 0x80000000u) ? (u & 0x7FFFFFFFu) : ~u);
}

// ---------------------------------------------------------------------------
// K0: zero-init agg / denom, emax = 0 (ordered-min, below every mapped float)
// ---------------------------------------------------------------------------
__global__ void init_kernel(unsigned* __restrict__ emax, float* __restrict__ denom,
                            float* __restrict__ agg, int n_agg, int n_nh) {
    int i = blockIdx.x * blockDim.x + threadIdx.x;
    if (i < n_agg) agg[i] = 0.0f;
    if (i < n_nh) { emax[i] = 0u; denom[i] = 0.0f; }
}

// ---------------------------------------------------------------------------
// K1: h = x @ emb_w + emb_b   (K=9, VALU; weights staged in LDS)
// ---------------------------------------------------------------------------
__global__ void emb_kernel(const float* __restrict__ x, const float* __restrict__ ew,
                           const float* __restrict__ eb, float* __restrict__ h, int n) {
    __shared__ float sw[9 * 128];
    __shared__ float sb[128];
    for (int i = threadIdx.x; i < 9 * 128; i += 256) sw[i] = ew[i];
    if (threadIdx.x < 128) sb[threadIdx.x] = eb[threadIdx.x];
    __syncthreads();
    int node = blockIdx.x * 2 + (threadIdx.x >> 7);
    int c = threadIdx.x & 127;
    if (node >= n) return;
    const float* xr = x + (size_t)node * 9;
    float acc = sb[c];
#pragma unroll
    for (int k = 0; k < 9; ++k) acc = fmaf(xr[k], sw[k * 128 + c], acc);
    h[(size_t)node * 128 + c] = acc;
}

// ---------------------------------------------------------------------------
// K2: xh = h @ gat_w via V_WMMA_F32_16X16X4_F32 (exact fp32).
// Block = 256 threads = 8 waves. Block covers one 16-row M tile; wave w covers
// N-tile (blockIdx.y*8 + w)*16. K=128 in 32 steps of 4.
// A tile (16x128 fp32, contiguous rows of h) staged in LDS.
// ISA 32-bit layouts:
//   A 16x4 : lane m=L%16; lanes<16 hold K=k0,k0+1; lanes>=16 hold K=k0+2,k0+3
//   B  4x16: lane n=L%16; same K split across lane halves
//   C/D    : vgpr g -> M = g + (L/16)*8, N = L%16
// ---------------------------------------------------------------------------
__global__ void __launch_bounds__(256)
gemm_xh_wmma(const float* __restrict__ h, const float* __restrict__ gw,
             float* __restrict__ xh, int n) {
    __shared__ float lds_a[16 * KDIM];   // 8 KB

    const int m0 = blockIdx.x * 16;
    // cooperative coalesced load: 16 rows of h are one contiguous 2048-float block
    for (int i = threadIdx.x; i < 16 * KDIM; i += 256) {
        int gidx = m0 * KDIM + i;
        lds_a[i] = (gidx < n * KDIM) ? h[gidx] : 0.0f;
    }
    __syncthreads();

    const int wave  = threadIdx.x >> 5;
    const int lane  = threadIdx.x & 31;
    const int n0    = (blockIdx.y * 8 + wave) * 16;   // 24 N-tiles total
    const int row   = lane & 15;                      // M (for A) / N (for B)
    const int khalf = (lane >> 4) * 2;                // 0 or 2

    v8f acc = {};
#pragma unroll 4
    for (int k0 = 0; k0 < KDIM; k0 += 4) {
        v2f a, b;
        const float* ap = &lds_a[row * KDIM + k0 + khalf];
        a.x = ap[0];
        a.y = ap[1];
        const float* bp = gw + (size_t)(k0 + khalf) * HC + n0 + row;
        b.x = bp[0];
        b.y = bp[HC];
        acc = __builtin_amdgcn_wmma_f32_16x16x4_f32(
            /*neg_a=*/false, a, /*neg_b=*/false, b,
            /*c_mod=*/(short)0, acc, /*reuse_a=*/false, /*reuse_b=*/false);
    }

    const int mbase = (lane >> 4) * 8;
#pragma unroll
    for (int g = 0; g < 8; ++g) {
        int m = m0 + mbase + g;
        if (m < n) xh[(size_t)m * HC + n0 + row] = acc[g];
    }
}

// ---------------------------------------------------------------------------
// K3: a_src/a_dst = einsum('nhc,hc->nh'). One wave per (node, head).
// ---------------------------------------------------------------------------
__global__ void att_kernel(const float* __restrict__ xh, const float* __restrict__ ws,
                           const float* __restrict__ wd, float* __restrict__ a_src,
                           float* __restrict__ a_dst, int n) {
    int gw = (blockIdx.x * blockDim.x + threadIdx.x) >> 5;
    int lane = threadIdx.x & 31;
    if (gw >= n * HEADS) return;
    int node = gw / HEADS, hd = gw % HEADS;
    const float* xr = xh + (size_t)node * HC + hd * CH;
    const float* sr = ws + hd * CH;
    const float* dr = wd + hd * CH;
    float ss = 0.0f, sd = 0.0f;
#pragma unroll
    for (int i = 0; i < 4; ++i) {
        int c = lane + i * 32;
        float v = xr[c];
        ss = fmaf(v, sr[c], ss);
        sd = fmaf(v, dr[c], sd);
    }
#pragma unroll
    for (int off = 16; off; off >>= 1) {
        ss += __shfl_xor(ss, off, 32);
        sd += __shfl_xor(sd, off, 32);
    }
    if (lane == 0) {
        a_src[node * HEADS + hd] = ss;
        a_dst[node * HEADS + hd] = sd;
    }
}

// ---------------------------------------------------------------------------
// K4: segment max over destination via ordered-uint atomic max (native).
// Thread per edge (incl. appended self loops), loop over heads.
// ---------------------------------------------------------------------------
__global__ void emax_kernel(const int* __restrict__ src, const int* __restrict__ dst,
                            const float* __restrict__ a_src, const float* __restrict__ a_dst,
                            unsigned* __restrict__ emax, int E, int n) {
    int t = blockIdx.x * blockDim.x + threadIdx.x;
    if (t >= E + n) return;
    int s, d;
    if (t < E) { s = src[t]; d = dst[t]; } else { s = d = t - E; }
#pragma unroll
    for (int hd = 0; hd < HEADS; ++hd) {
        float e = a_src[s * HEADS + hd] + a_dst[d * HEADS + hd];
        e = (e > 0.0f) ? e : NEG_SLOPE * e;
        atomicMax(&emax[d * HEADS + hd], f2ord(e));
    }
}

// ---------------------------------------------------------------------------
// K5: w = exp(e - emax[dst]); denom += w (native f32 atomic add); cache w.
// ---------------------------------------------------------------------------
__global__ void expden_kernel(const int* __restrict__ src, const int* __restrict__ dst,
                              const float* __restrict__ a_src, const float* __restrict__ a_dst,
                              const unsigned* __restrict__ emax, float* __restrict__ denom,
                              float* __restrict__ wbuf, int E, int n) {
    int t = blockIdx.x * blockDim.x + threadIdx.x;
    if (t >= E + n) return;
    int s, d;
    if (t < E) { s = src[t]; d = dst[t]; } else { s = d = t - E; }
#pragma unroll
    for (int hd = 0; hd < HEADS; ++hd) {
        float e = a_src[s * HEADS + hd] + a_dst[d * HEADS + hd];
        e = (e > 0.0f) ? e : NEG_SLOPE * e;
        float w = expf(e - ord2f(emax[d * HEADS + hd]));
        wbuf[t * HEADS + hd] = w;
        atomicAdd(&denom[d * HEADS + hd], w);
    }
}

// ---------------------------------------------------------------------------
// K6: agg[dst] += xh[src] * alpha. One wave per edge; lanes cover the 128
// channels of each head with float4 (coalesced 1536B gather per edge row).
// ---------------------------------------------------------------------------
__global__ void agg_kernel(const int* __restrict__ src, const int* __restrict__ dst,
                           const float* __restrict__ xh, const float* __restrict__ wbuf,
                           const float* __restrict__ denom, float* __restrict__ agg,
                           int E, int n) {
    int gw = (blockIdx.x * blockDim.x + threadIdx.x) >> 5;
    int lane = threadIdx.x & 31;
    if (gw >= E + n) return;
    int s, d;
    if (gw < E) { s = src[gw]; d = dst[gw]; } else { s = d = gw - E; }
    float alpha[HEADS];
#pragma unroll
    for (int hd = 0; hd < HEADS; ++hd)
        alpha[hd] = wbuf[gw * HEADS + hd] / (denom[d * HEADS + hd] + EPS);
    const float* xr = xh + (size_t)s * HC;
    float* ar = agg + (size_t)d * HC;
#pragma unroll
    for (int hd = 0; hd < HEADS; ++hd) {
        int off = hd * CH + lane * 4;
        const float4 v = *(const float4*)(xr + off);
        float a = alpha[hd];
        atomicAdd(ar + off + 0, v.x * a);
        atomicAdd(ar + off + 1, v.y * a);
        atomicAdd(ar + off + 2, v.z * a);
        atomicAdd(ar + off + 3, v.w * a);
    }
}

// ---------------------------------------------------------------------------
// K7: out = ((agg.mean(heads) + gat_bias + h) @ out_w) + out_b. Wave per node.
// ---------------------------------------------------------------------------
__global__ void final_kernel(const float* __restrict__ agg, const float* __restrict__ h,
                             const float* __restrict__ gbias, const float* __restrict__ ow,
                             const float* __restrict__ ob, float* __restrict__ out, int n) {
    int gw = (blockIdx.x * blockDim.x + threadIdx.x) >> 5;
    int lane = threadIdx.x & 31;
    if (gw >= n) return;
    const float* ar = agg + (size_t)gw * HC;
    const float* hr = h + (size_t)gw * CH;
    float p0 = 0.0f, p1 = 0.0f, p2 = 0.0f, p3 = 0.0f;
#pragma unroll
    for (int i = 0; i < 4; ++i) {
        int c = lane + i * 32;
        float xn = (ar[c] + ar[CH + c] + ar[2 * CH + c]) * (1.0f / 3.0f)
                   + gbias[c] + hr[c];
        p0 = fmaf(xn, ow[c * 4 + 0], p0);
        p1 = fmaf(xn, ow[c * 4 + 1], p1);
        p2 = fmaf(xn, ow[c * 4 + 2], p2);
        p3 = fmaf(xn, ow[c * 4 + 3], p3);
    }
#pragma unroll
    for (int off = 16; off; off >>= 1) {
        p0 += __shfl_xor(p0, off, 32);
        p1 += __shfl_xor(p1, off, 32);
        p2 += __shfl_xor(p2, off, 32);
        p3 += __shfl_xor(p3, off, 32);
    }
    if (lane == 0) {
        out[gw * 4 + 0] = p0 + ob[0];
        out[gw * 4 + 1] = p1 + ob[1];
        out[gw * 4 + 2] = p2 + ob[2];
        out[gw * 4 + 3] = p3 + ob[3];
    }
}

// ---------------------------------------------------------------------------
extern "C" void kernel_launch(void* const* d_in, const int* in_sizes, int n_in,
                              void* d_out, int out_size, void* d_ws, size_t ws_size,
                              hipStream_t stream) {
    const float* x     = (const float*)d_in[0];
    const float* emb_w = (const float*)d_in[1];
    const float* emb_b = (const float*)d_in[2];
    const float* gat_w = (const float*)d_in[3];
    const float* att_s = (const float*)d_in[4];
    const float* att_d = (const float*)d_in[5];
    const float* gbias = (const float*)d_in[6];
    const float* out_w = (const float*)d_in[7];
    const float* out_b = (const float*)d_in[8];
    const int*   eidx  = (const int*)d_in[9];
    float* out = (float*)d_out;

    const int n    = in_sizes[0] / 9;   // 50000
    const int E    = in_sizes[9] / 2;   // 400000
    const int Etot = E + n;             // edges + self loops
    const int* src = eidx;
    const int* dst = eidx + E;

    // Workspace layout (floats). Total ~ 46.75M floats ~= 187 MB.
    float*    h     = (float*)d_ws;                       // n*128
    float*    xh    = h     + (size_t)n * CH;             // n*384
    float*    a_src = xh    + (size_t)n * HC;             // n*3
    float*    a_dst = a_src + (size_t)n * HEADS;          // n*3
    unsigned* emax  = (unsigned*)(a_dst + (size_t)n * HEADS); // n*3
    float*    denom = (float*)(emax + (size_t)n * HEADS); // n*3
    float*    wbuf  = denom + (size_t)n * HEADS;          // Etot*3
    float*    agg   = wbuf  + (size_t)Etot * HEADS;       // n*384

    const int n_agg = n * HC;
    init_kernel<<<(n_agg + 255) / 256, 256, 0, stream>>>(emax, denom, agg, n_agg, n * HEADS);

    emb_kernel<<<(n + 1) / 2, 256, 0, stream>>>(x, emb_w, emb_b, h, n);

    gemm_xh_wmma<<<dim3((n + 15) / 16, HC / (8 * 16)), 256, 0, stream>>>(h, gat_w, xh, n);

    att_kernel<<<(n * HEADS * 32 + 255) / 256, 256, 0, stream>>>(xh, att_s, att_d, a_src, a_dst, n);

    emax_kernel<<<(Etot + 255) / 256, 256, 0, stream>>>(src, dst, a_src, a_dst, emax, E, n);

    expden_kernel<<<(Etot + 255) / 256, 256, 0, stream>>>(src, dst, a_src, a_dst, emax, denom, wbuf, E, n);

    agg_kernel<<<(Etot * 32 + 255) / 256, 256, 0, stream>>>(src, dst, xh, wbuf, denom, agg, E, n);

    final_kernel<<<((size_t)n * 32 + 255) / 256, 256, 0, stream>>>(agg, h, gbias, out_w, out_b, out, n);
}